// RGCNEncoder_6545530159457
// MI455X (gfx1250) — compile-verified
//
#include <hip/hip_runtime.h>

typedef __attribute__((ext_vector_type(2))) float v2f;
typedef __attribute__((ext_vector_type(8))) float v8f;

#define H 128
#define EPS 1e-5f

__device__ __forceinline__ v8f wmma_f32(v2f a, v2f b, v8f c) {
    // V_WMMA_F32_16X16X4_F32 : D = A(16x4) * B(4x16) + C(16x16), all fp32
    return __builtin_amdgcn_wmma_f32_16x16x4_f32(
        /*neg_a=*/false, a, /*neg_b=*/false, b,
        /*c_mod=*/(short)0, c, /*reuse_a=*/false, /*reuse_b=*/false);
}

// ---------------------------------------------------------------------------
// out[M x 128] = relu(X[M x K] @ W[128 x K]^T + bias)
// 256 threads = 8 waves; wave handles 16 rows; block handles 128 rows.
// ---------------------------------------------------------------------------
template <int K>
__global__ __launch_bounds__(256) void gemm_relu(
    const float* __restrict__ X, const float* __restrict__ W,
    const float* __restrict__ bias, float* __restrict__ out, int M) {
    const int lane  = threadIdx.x & 31;
    const int wave  = threadIdx.x >> 5;
    const int row0  = blockIdx.x * 128 + wave * 16;
    const int l16   = lane & 15;
    const int hi    = lane >> 4;            // 0 for lanes 0-15, 1 for 16-31
    const int arow  = min(row0 + l16, M - 1);
    const int khalf = hi * 2;               // A/B K sub-offset per half-wave

    v8f acc[8];
#pragma unroll
    for (int t = 0; t < 8; ++t) acc[t] = (v8f){0.f,0.f,0.f,0.f,0.f,0.f,0.f,0.f};

    for (int k0 = 0; k0 < K; k0 += 4) {
        v2f a = *(const v2f*)(X + (size_t)arow * K + k0 + khalf);
#pragma unroll
        for (int t = 0; t < 8; ++t) {
            const int n = t * 16 + l16;
            v2f b = *(const v2f*)(W + (size_t)n * K + k0 + khalf);
            acc[t] = wmma_f32(a, b, acc[t]);
        }
    }

#pragma unroll
    for (int t = 0; t < 8; ++t) {
        const int n  = t * 16 + l16;
        const float bn = bias[n];
#pragma unroll
        for (int r = 0; r < 8; ++r) {
            const int row = row0 + r + (hi << 3);
            if (row < M) {
                float v = acc[t][r] + bn;
                out[(size_t)row * H + n] = fmaxf(v, 0.f);
            }
        }
    }
}

// ---------------------------------------------------------------------------
// Edge scatter for segment-mean: one wave per edge, float4 per lane.
// ---------------------------------------------------------------------------
__global__ __launch_bounds__(256) void scatter_add(
    const float* __restrict__ hsrc, const int* __restrict__ ei, int n_edges,
    float* __restrict__ acc, float* __restrict__ cnt) {
    const int edge = (int)((blockIdx.x * (size_t)blockDim.x + threadIdx.x) >> 5);
    if (edge >= n_edges) return;
    const int lane = threadIdx.x & 31;
    const int src  = ei[edge];
    const int dst  = ei[n_edges + edge];
    const float4 v = *(const float4*)(hsrc + (size_t)src * H + lane * 4);
    float* a = acc + (size_t)dst * H + lane * 4;
    atomicAdd(a + 0, v.x);
    atomicAdd(a + 1, v.y);
    atomicAdd(a + 2, v.z);
    atomicAdd(a + 3, v.w);
    if (lane == 0) atomicAdd(cnt + dst, 1.0f);
}

// ---------------------------------------------------------------------------
// out = LN( hdst + (acc/max(cnt,1)) @ wl^T + bl + hdst @ wr^T ) [, relu]
// ---------------------------------------------------------------------------
__global__ __launch_bounds__(256) void sage_ln(
    const float* __restrict__ acc, const float* __restrict__ cnt,
    const float* __restrict__ hdst,
    const float* __restrict__ wl, const float* __restrict__ bl,
    const float* __restrict__ wr,
    const float* __restrict__ g, const float* __restrict__ b,
    float* __restrict__ out, int M, int do_relu) {
    const int lane  = threadIdx.x & 31;
    const int wave  = threadIdx.x >> 5;
    const int row0  = blockIdx.x * 128 + wave * 16;
    const int l16   = lane & 15;
    const int hi    = lane >> 4;
    const int arow  = min(row0 + l16, M - 1);
    const int khalf = hi * 2;
    const float rc  = 1.0f / fmaxf(cnt[arow], 1.0f);

    v8f accv[8];
#pragma unroll
    for (int t = 0; t < 8; ++t) accv[t] = (v8f){0.f,0.f,0.f,0.f,0.f,0.f,0.f,0.f};

    for (int k0 = 0; k0 < H; k0 += 4) {
        v2f a1 = *(const v2f*)(acc  + (size_t)arow * H + k0 + khalf);
        a1 = a1 * rc;                                   // mean aggregation
        v2f a2 = *(const v2f*)(hdst + (size_t)arow * H + k0 + khalf);
#pragma unroll
        for (int t = 0; t < 8; ++t) {
            const int n = t * 16 + l16;
            v2f b1 = *(const v2f*)(wl + (size_t)n * H + k0 + khalf);
            accv[t] = wmma_f32(a1, b1, accv[t]);
            v2f b2 = *(const v2f*)(wr + (size_t)n * H + k0 + khalf);
            accv[t] = wmma_f32(a2, b2, accv[t]);
        }
    }

    float blv[8], gv[8], bv[8];
#pragma unroll
    for (int t = 0; t < 8; ++t) {
        const int n = t * 16 + l16;
        blv[t] = bl[n]; gv[t] = g[n]; bv[t] = b[n];
    }

#pragma unroll
    for (int r = 0; r < 8; ++r) {
        const int rowr = row0 + r + (hi << 3);
        const int rowc = min(rowr, M - 1);
        float xv[8];
        float s = 0.f, s2 = 0.f;
#pragma unroll
        for (int t = 0; t < 8; ++t) {
            const int n = t * 16 + l16;
            const float v = accv[t][r] + blv[t] + hdst[(size_t)rowc * H + n];
            xv[t] = v; s += v; s2 += v * v;
        }
        // row M=r lives in lanes 0-15, row M=r+8 in lanes 16-31:
        // xor-masks 8,4,2,1 reduce each 16-lane half independently (wave32)
#pragma unroll
        for (int m = 8; m >= 1; m >>= 1) {
            s  += __shfl_xor(s,  m, 32);
            s2 += __shfl_xor(s2, m, 32);
        }
        const float mu  = s * (1.0f / H);
        const float var = s2 * (1.0f / H) - mu * mu;
        const float rs  = rsqrtf(fmaxf(var, 0.f) + EPS);
        if (rowr < M) {
#pragma unroll
            for (int t = 0; t < 8; ++t) {
                const int n = t * 16 + l16;
                float y = (xv[t] - mu) * rs * gv[t] + bv[t];
                if (do_relu) y = fmaxf(y, 0.f);
                out[(size_t)rowr * H + n] = y;
            }
        }
    }
}

// ---------------------------------------------------------------------------
extern "C" void kernel_launch(void* const* d_in, const int* in_sizes, int n_in,
                              void* d_out, int out_size, void* d_ws, size_t ws_size,
                              hipStream_t stream) {
    const float* x_user = (const float*)d_in[0];
    const float* x_item = (const float*)d_in[1];
    const int*   ei_rates = (const int*)d_in[2];
    const int*   ei_rev   = (const int*)d_in[3];
    const float* pu_w = (const float*)d_in[4];
    const float* pu_b = (const float*)d_in[5];
    const float* pi_w = (const float*)d_in[6];
    const float* pi_b = (const float*)d_in[7];
    const float* lnu_g = (const float*)d_in[8];
    const float* lnu_b = (const float*)d_in[9];
    const float* lni_g = (const float*)d_in[10];
    const float* lni_b = (const float*)d_in[11];
    // per (layer, relation): wl, bl, wr   starting at index 12
    const float* L[2][2][3];
    for (int l = 0; l < 2; ++l)
        for (int rel = 0; rel < 2; ++rel)
            for (int w = 0; w < 3; ++w)
                L[l][rel][w] = (const float*)d_in[12 + l * 6 + rel * 3 + w];

    const int N_user = in_sizes[0] / 256;
    const int N_item = in_sizes[1] / 256;
    const int nE_rates = in_sizes[2] / 2;
    const int nE_rev   = in_sizes[3] / 2;

    const size_t NHu = (size_t)N_user * H;
    const size_t NHi = (size_t)N_item * H;
    float* ws   = (float*)d_ws;
    float* hu0  = ws;
    float* hi0  = hu0 + NHu;
    float* hu1  = hi0 + NHi;
    float* hi1  = hu1 + NHu;
    float* acc_u = hi1 + NHi;
    float* acc_i = acc_u + NHu;
    float* cnt_u = acc_i + NHi;
    float* cnt_i = cnt_u + N_user;

    float* hu_out = (float*)d_out;            // [0] of stack
    float* hi_out = hu_out + NHu;             // [1] of stack

    const dim3 blk(256);
    const dim3 grd_u((N_user + 127) / 128);
    const dim3 grd_i((N_item + 127) / 128);
    const dim3 grd_e1((nE_rates + 7) / 8);
    const dim3 grd_e2((nE_rev + 7) / 8);

    // input projection + relu
    gemm_relu<256><<<grd_u, blk, 0, stream>>>(x_user, pu_w, pu_b, hu0, N_user);
    gemm_relu<256><<<grd_i, blk, 0, stream>>>(x_item, pi_w, pi_b, hi0, N_item);

    const size_t acc_bytes = (NHu + NHi + (size_t)N_user + (size_t)N_item) * sizeof(float);

    // ---- layer 0 (relu at end) ----
    hipMemsetAsync(acc_u, 0, acc_bytes, stream);
    scatter_add<<<grd_e1, blk, 0, stream>>>(hu0, ei_rates, nE_rates, acc_i, cnt_i);
    scatter_add<<<grd_e2, blk, 0, stream>>>(hi0, ei_rev,   nE_rev,   acc_u, cnt_u);
    sage_ln<<<grd_i, blk, 0, stream>>>(acc_i, cnt_i, hi0,
        L[0][0][0], L[0][0][1], L[0][0][2], lni_g, lni_b, hi1, N_item, 1);
    sage_ln<<<grd_u, blk, 0, stream>>>(acc_u, cnt_u, hu0,
        L[0][1][0], L[0][1][1], L[0][1][2], lnu_g, lnu_b, hu1, N_user, 1);

    // ---- layer 1 (no relu, write straight to output) ----
    hipMemsetAsync(acc_u, 0, acc_bytes, stream);
    scatter_add<<<grd_e1, blk, 0, stream>>>(hu1, ei_rates, nE_rates, acc_i, cnt_i);
    scatter_add<<<grd_e2, blk, 0, stream>>>(hi1, ei_rev,   nE_rev,   acc_u, cnt_u);
    sage_ln<<<grd_i, blk, 0, stream>>>(acc_i, cnt_i, hi1,
        L[1][0][0], L[1][0][1], L[1][0][2], lni_g, lni_b, hi_out, N_item, 0);
    sage_ln<<<grd_u, blk, 0, stream>>>(acc_u, cnt_u, hu1,
        L[1][1][0], L[1][1][1], L[1][1][2], lnu_g, lnu_b, hu_out, N_user, 0);
}